// GCN1_3745211482881
// MI455X (gfx1250) — compile-verified
//
#include <hip/hip_runtime.h>
#include <hip/hip_bf16.h>

#define N_NODES 15828
#define BATCH   64
#define N_EDGES 253248
#define HDIM    100
#define HPAD    112      // 7 tiles of 16
#define NCLS    10
#define NEG     0.01f
#define NB      (N_NODES * BATCH)

typedef __attribute__((ext_vector_type(2))) float v2f;
typedef __attribute__((ext_vector_type(8))) float v8f;

__device__ __forceinline__ float leaky(float x) { return x > 0.f ? x : NEG * x; }

// ---------------- utility ----------------
__global__ void k_zero(float* __restrict__ p, int n) {
    int i = blockIdx.x * blockDim.x + threadIdx.x;
    if (i < n) p[i] = 0.f;
}

// ---------------- degrees ----------------
__global__ void k_degree(const int* __restrict__ src, const int* __restrict__ dst,
                         float* __restrict__ cnt_out, float* __restrict__ cnt_in) {
    int e = blockIdx.x * blockDim.x + threadIdx.x;
    if (e < N_EDGES) {
        atomicAdd(&cnt_out[src[e]], 1.f);
        atomicAdd(&cnt_in[dst[e]], 1.f);
    }
}

__global__ void k_rsqrt_clip(float* __restrict__ p, int n) {
    int i = blockIdx.x * blockDim.x + threadIdx.x;
    if (i < n) p[i] = rsqrtf(fmaxf(p[i], 1.f));
}

// ---------------- conv0 front: src-normalize ----------------
__global__ void k_x0(const float* __restrict__ in_feat, const float* __restrict__ dinv_out,
                     float* __restrict__ x0) {
    int i = blockIdx.x * blockDim.x + threadIdx.x;
    if (i < NB) {
        int n = i >> 6;                       // BATCH == 64
        x0[i] = in_feat[i] * dinv_out[n];
    }
}

// ---------------- edge gather + scatter-add (used twice) ----------------
__global__ void k_scatter(const int* __restrict__ src, const int* __restrict__ dst,
                          const float* __restrict__ x, float* __restrict__ agg) {
    int i = blockIdx.x * blockDim.x + threadIdx.x;
    if (i < N_EDGES * BATCH) {
        int e = i >> 6;
        int b = i & 63;
        int s = src[e];
        int d = dst[e];
        atomicAdd(&agg[d * BATCH + b], x[s * BATCH + b]);
    }
}

// ---------------- fused: dst-norm, W0 expand, bias, leaky, W1 contract, src-norm ----------------
__global__ void k_x1(float* __restrict__ agg1_x1,     // in-place [N,B]
                     const float* __restrict__ dinv_in, const float* __restrict__ dinv_out,
                     const float* __restrict__ W0, const float* __restrict__ b0,
                     const float* __restrict__ W1) {
    int i = blockIdx.x * blockDim.x + threadIdx.x;
    if (i < NB) {
        int n = i >> 6;
        float a = agg1_x1[i] * dinv_in[n];
        float acc = 0.f;
#pragma unroll 4
        for (int f = 0; f < HDIM; ++f) {
            float v = fmaf(a, W0[f], b0[f]);
            v = leaky(v);
            acc = fmaf(v, W1[f], acc);
        }
        agg1_x1[i] = acc * dinv_out[n];
    }
}

// ---------------- conv1 tail: dst-norm + bias + leaky, transpose to [B,N] ----------------
__global__ void k_yT(const float* __restrict__ agg2, const float* __restrict__ dinv_in,
                     const float* __restrict__ b1, float* __restrict__ yT) {
    int i = blockIdx.x * blockDim.x + threadIdx.x;
    if (i < NB) {
        int n = i >> 6;
        int b = i & 63;
        yT[b * N_NODES + n] = leaky(fmaf(agg2[i], dinv_in[n], b1[0]));
    }
}

// ---------------- WMMA GEMM1: Z1[64,HPAD] = leaky(yT[64,N] @ lw0^T[N,100] + lb0) ----------------
// 4 M-tiles x 7 N-tiles, one wave32 per tile, V_WMMA_F32_16X16X4_F32 over K=N_NODES.
__global__ void k_gemm1(const float* __restrict__ yT, const float* __restrict__ lw0,
                        const float* __restrict__ lb0, float* __restrict__ Z1) {
    int lane = threadIdx.x;           // 0..31
    int tile = blockIdx.x;            // 0..27
    int mt = tile / 7, nt = tile % 7;
    int half = lane >> 4;             // 0 or 1
    int l16  = lane & 15;
    int arow = mt * 16 + l16;         // M index for A fragment (<64 always)
    int bcol = nt * 16 + l16;         // N index for B fragment (may exceed 99)
    int kbase = 2 * half;             // VGPR0 -> K={0,2}, VGPR1 -> K={1,3}

    int  bc    = bcol < HDIM ? bcol : (HDIM - 1);
    float bmsk = bcol < HDIM ? 1.f : 0.f;
    const float* ap = yT  + (long long)arow * N_NODES;
    const float* bp = lw0 + (long long)bc   * N_NODES;

    v8f c = {0.f, 0.f, 0.f, 0.f, 0.f, 0.f, 0.f, 0.f};
    for (int k0 = 0; k0 < N_NODES; k0 += 4) {
        v2f a, b;
        a.x = ap[k0 + kbase];
        a.y = ap[k0 + kbase + 1];
        b.x = bp[k0 + kbase] * bmsk;
        b.y = bp[k0 + kbase + 1] * bmsk;
        c = __builtin_amdgcn_wmma_f32_16x16x4_f32(false, a, false, b, (short)0, c,
                                                  false, false);
    }
    float bias = (bcol < HDIM) ? lb0[bc] : 0.f;
#pragma unroll
    for (int i = 0; i < 8; ++i) {
        int row = mt * 16 + i + 8 * half;       // C/D layout: VGPR i -> M=i (+8 for hi half)
        Z1[row * HPAD + bcol] = leaky(c[i] + bias);
    }
}

// ---------------- WMMA GEMM2: Z2[64,HPAD] = leaky(Z1[64,HPAD] @ lw2^T + lb2) ----------------
__global__ void k_gemm2(const float* __restrict__ Z1, const float* __restrict__ lw2,
                        const float* __restrict__ lb2, float* __restrict__ Z2) {
    int lane = threadIdx.x;
    int tile = blockIdx.x;
    int mt = tile / 7, nt = tile % 7;
    int half = lane >> 4;
    int l16  = lane & 15;
    int arow = mt * 16 + l16;
    int bcol = nt * 16 + l16;
    int kbase = 2 * half;

    int bc = bcol < HDIM ? bcol : (HDIM - 1);
    const float* ap = Z1  + (long long)arow * HPAD;
    const float* bp = lw2 + (long long)bc   * HDIM;
    float fmsk = bcol < HDIM ? 1.f : 0.f;

    v8f c = {0.f, 0.f, 0.f, 0.f, 0.f, 0.f, 0.f, 0.f};
    for (int k0 = 0; k0 < HPAD; k0 += 4) {
        int ka = k0 + kbase;
        int kb = ka + 1;
        float m0 = (ka < HDIM) ? fmsk : 0.f;
        float m1 = (kb < HDIM) ? fmsk : 0.f;
        int ia = ka < HDIM ? ka : (HDIM - 1);
        int ib = kb < HDIM ? kb : (HDIM - 1);
        v2f a, b;
        a.x = ap[ka];                 // Z1 pad cols are exact zeros
        a.y = ap[kb];
        b.x = bp[ia] * m0;
        b.y = bp[ib] * m1;
        c = __builtin_amdgcn_wmma_f32_16x16x4_f32(false, a, false, b, (short)0, c,
                                                  false, false);
    }
    float bias = (bcol < HDIM) ? lb2[bc] : 0.f;
#pragma unroll
    for (int i = 0; i < 8; ++i) {
        int row = mt * 16 + i + 8 * half;
        Z2[row * HPAD + bcol] = leaky(c[i] + bias);
    }
}

// ---------------- head: out[B,10] = leaky(Z2[:, :100] @ lw3^T + lb3) ----------------
__global__ void k_out(const float* __restrict__ Z2, const float* __restrict__ lw3,
                      const float* __restrict__ lb3, float* __restrict__ out) {
    int i = blockIdx.x * blockDim.x + threadIdx.x;
    if (i < BATCH * NCLS) {
        int b = i / NCLS;
        int c = i % NCLS;
        float acc = lb3[c];
#pragma unroll 4
        for (int f = 0; f < HDIM; ++f)
            acc = fmaf(Z2[b * HPAD + f], lw3[c * HDIM + f], acc);
        out[i] = leaky(acc);
    }
}

extern "C" void kernel_launch(void* const* d_in, const int* in_sizes, int n_in,
                              void* d_out, int out_size, void* d_ws, size_t ws_size,
                              hipStream_t stream) {
    const float* in_feat = (const float*)d_in[0];
    const int*   eidx    = (const int*)d_in[1];
    const int*   src     = eidx;
    const int*   dst     = eidx + N_EDGES;
    const float* W0  = (const float*)d_in[2];
    const float* b0  = (const float*)d_in[3];
    const float* W1  = (const float*)d_in[4];
    const float* b1  = (const float*)d_in[5];
    const float* lw0 = (const float*)d_in[6];
    const float* lb0 = (const float*)d_in[7];
    const float* lw2 = (const float*)d_in[8];
    const float* lb2 = (const float*)d_in[9];
    const float* lw3 = (const float*)d_in[10];
    const float* lb3 = (const float*)d_in[11];
    float* out = (float*)d_out;

    float* ws       = (float*)d_ws;
    float* dinv_out = ws;                      // [N]
    float* dinv_in  = ws + N_NODES;            // [N]
    float* x0       = ws + 2 * N_NODES;        // [N,B], reused as yT [B,N]
    float* agg1     = x0 + NB;                 // [N,B], reused in-place as x1
    float* agg2     = agg1 + NB;               // [N,B]
    float* Z1       = agg2 + NB;               // [64,HPAD]
    float* Z2       = Z1 + BATCH * HPAD;       // [64,HPAD]

    const int T = 256;
    // re-initialize accumulators every call (graph-replay safe)
    k_zero<<<(2 * N_NODES + T - 1) / T, T, 0, stream>>>(dinv_out, 2 * N_NODES);
    k_zero<<<(2 * NB + T - 1) / T, T, 0, stream>>>(agg1, 2 * NB);

    k_degree<<<(N_EDGES + T - 1) / T, T, 0, stream>>>(src, dst, dinv_out, dinv_in);
    k_rsqrt_clip<<<(2 * N_NODES + T - 1) / T, T, 0, stream>>>(dinv_out, 2 * N_NODES);

    k_x0<<<(NB + T - 1) / T, T, 0, stream>>>(in_feat, dinv_out, x0);
    k_scatter<<<(N_EDGES * BATCH + T - 1) / T, T, 0, stream>>>(src, dst, x0, agg1);
    k_x1<<<(NB + T - 1) / T, T, 0, stream>>>(agg1, dinv_in, dinv_out, W0, b0, W1);
    k_scatter<<<(N_EDGES * BATCH + T - 1) / T, T, 0, stream>>>(src, dst, agg1, agg2);
    k_yT<<<(NB + T - 1) / T, T, 0, stream>>>(agg2, dinv_in, b1, x0);   // x0 := yT

    k_gemm1<<<28, 32, 0, stream>>>(x0, lw0, lb0, Z1);
    k_gemm2<<<28, 32, 0, stream>>>(Z1, lw2, lb2, Z2);
    k_out<<<(BATCH * NCLS + T - 1) / T, T, 0, stream>>>(Z2, lw3, lb3, out);
}